// MultiStepIFNode_45896020525771
// MI455X (gfx1250) — compile-verified
//
#include <hip/hip_runtime.h>

// Multi-step integrate-and-fire (SpikingJelly-style MultiStepIFNode), f32.
//   x_seq: (T=16, B, C, H, W) -> spike_seq same shape.
// Pure HBM-streaming problem: 512 MiB total traffic, ~0.4 FLOP/byte.
// Strategy: one lane owns 4 consecutive spatial cells (float4), hoists all
// T=16 non-temporal b128 loads (8 KB in flight per wave32), runs the trivial
// serial IF recurrence in registers, and streams NT b128 stores back out.

typedef float v4f __attribute__((ext_vector_type(4)));

#ifndef IF_T
#define IF_T 16
#endif

__global__ __launch_bounds__(256) void if_multistep_kernel(
    const float* __restrict__ x, float* __restrict__ out, long long n4) {
  long long i = (long long)blockIdx.x * blockDim.x + threadIdx.x;
  if (i >= n4) return;

  const v4f* __restrict__ xp = reinterpret_cast<const v4f*>(x) + i;
  v4f* __restrict__ op = reinterpret_cast<v4f*>(out) + i;

  // Issue all T coalesced non-temporal b128 loads up front: 16 outstanding
  // LOADcnt entries per lane; stride between timesteps is n4 float4's.
  v4f xv[IF_T];
#pragma unroll
  for (int t = 0; t < IF_T; ++t) {
    xv[t] = __builtin_nontemporal_load(xp + (long long)t * n4);
  }

  const float th = 1.0f;  // V_THRESHOLD
  v4f v = {0.0f, 0.0f, 0.0f, 0.0f};

#pragma unroll
  for (int t = 0; t < IF_T; ++t) {
    v4f s;
    v.x += xv[t].x; s.x = (v.x >= th) ? 1.0f : 0.0f; v.x -= s.x * th;
    v.y += xv[t].y; s.y = (v.y >= th) ? 1.0f : 0.0f; v.y -= s.y * th;
    v.z += xv[t].z; s.z = (v.z >= th) ? 1.0f : 0.0f; v.z -= s.z * th;
    v.w += xv[t].w; s.w = (v.w >= th) ? 1.0f : 0.0f; v.w -= s.w * th;
    // Write-once output: non-temporal b128 store, don't pollute L2.
    __builtin_nontemporal_store(s, op + (long long)t * n4);
  }
}

extern "C" void kernel_launch(void* const* d_in, const int* in_sizes, int n_in,
                              void* d_out, int out_size, void* d_ws, size_t ws_size,
                              hipStream_t stream) {
  (void)n_in; (void)out_size; (void)d_ws; (void)ws_size;

  const float* x = reinterpret_cast<const float*>(d_in[0]);
  float* out = reinterpret_cast<float*>(d_out);

  const long long total = (long long)in_sizes[0];   // T * B * C * H * W
  const long long nspatial = total / IF_T;          // B * C * H * W
  const long long n4 = nspatial / 4;                // float4 lanes

  const int block = 256;  // 8 wave32's per block
  const long long grid = (n4 + block - 1) / block;

  if_multistep_kernel<<<dim3((unsigned)grid), dim3(block), 0, stream>>>(x, out, n4);
}